// SetAbstractionLayer_71330816852083
// MI455X (gfx1250) — compile-verified
//
#include <hip/hip_runtime.h>
#include <hip/hip_bf16.h>

#define B_   8
#define N_   4096
#define C_   32
#define S_   1024
#define K_   64
#define OUT_ 64
#define R2_  0.04f

typedef float v2f __attribute__((ext_vector_type(2)));
typedef float v8f __attribute__((ext_vector_type(8)));

__device__ __forceinline__ void argmax_combine(float& bv, int& bi, float ov, int oi) {
  // strictly-greater wins; on ties keep the LOWER index (jnp.argmax semantics)
  if (ov > bv || (ov == bv && oi < bi)) { bv = ov; bi = oi; }
}

// ---------------------------------------------------------------------------
// Kernel 1: farthest-point sampling. One block (1024 thr) per batch.
// LDS: pos SoA (48KB) + mind (16KB) + reduce scratch.
// ---------------------------------------------------------------------------
__global__ void fps_kernel(const float* __restrict__ pos, float* __restrict__ cent) {
  extern __shared__ float sm[];
  float* px   = sm;
  float* py   = px + N_;
  float* pz   = py + N_;
  float* mind = pz + N_;
  float* rv   = mind + N_;            // 32 wave maxima
  int*   ri   = (int*)(rv + 32);      // 32 wave argmax indices
  const int b = blockIdx.x;
  const int t = threadIdx.x;          // 0..1023
  const float* pb = pos + (size_t)b * N_ * 3;

  for (int i = t; i < N_; i += 1024) {
    px[i] = pb[i * 3 + 0];
    py[i] = pb[i * 3 + 1];
    pz[i] = pb[i * 3 + 2];
  }
  __syncthreads();

  const float x0 = px[0], y0 = py[0], z0 = pz[0];
  for (int i = t; i < N_; i += 1024) {
    float dx = px[i] - x0, dy = py[i] - y0, dz = pz[i] - z0;
    mind[i] = dx * dx + dy * dy + dz * dz;
  }
  if (t == 0) {
    cent[(b * S_ + 0) * 3 + 0] = x0;
    cent[(b * S_ + 0) * 3 + 1] = y0;
    cent[(b * S_ + 0) * 3 + 2] = z0;
  }
  __syncthreads();

  const int wave = t >> 5, lane = t & 31;
  for (int s = 1; s < S_; ++s) {
    float bv = -1.0f; int bi = 0x7fffffff;
    for (int i = t; i < N_; i += 1024) argmax_combine(bv, bi, mind[i], i);
    #pragma unroll
    for (int off = 16; off > 0; off >>= 1) {
      float ov = __shfl_down(bv, off, 32);
      int   oi = __shfl_down(bi, off, 32);
      argmax_combine(bv, bi, ov, oi);
    }
    if (lane == 0) { rv[wave] = bv; ri[wave] = bi; }
    __syncthreads();
    if (wave == 0) {
      bv = rv[lane]; bi = ri[lane];
      #pragma unroll
      for (int off = 16; off > 0; off >>= 1) {
        float ov = __shfl_down(bv, off, 32);
        int   oi = __shfl_down(bi, off, 32);
        argmax_combine(bv, bi, ov, oi);
      }
      if (lane == 0) ri[0] = bi;
    }
    __syncthreads();
    const int idx = ri[0];
    const float cx = px[idx], cy = py[idx], cz = pz[idx];
    if (t == 0) {
      cent[(b * S_ + s) * 3 + 0] = cx;
      cent[(b * S_ + s) * 3 + 1] = cy;
      cent[(b * S_ + s) * 3 + 2] = cz;
    }
    for (int i = t; i < N_; i += 1024) {
      float dx = px[i] - cx, dy = py[i] - cy, dz = pz[i] - cz;
      float d = dx * dx + dy * dy + dz * dz;
      mind[i] = fminf(mind[i], d);
    }
    __syncthreads();
  }
}

// ---------------------------------------------------------------------------
// Kernel 2: per-point MLP via exact-f32 WMMA (V_WMMA_F32_16X16X4_F32).
// 256 thr = 8 waves; each wave owns a 16-row M tile; block covers 128 rows.
// f32 A layout: lanes 0-15 hold K={0,1}, lanes 16-31 hold K={2,3}.
// B/C/D: row striped across lanes within a VGPR; VGPR v = rows {v, v+rows/2}.
// ---------------------------------------------------------------------------
__global__ void mlp_wmma_kernel(const float* __restrict__ h, const float* __restrict__ pos,
                                const float* __restrict__ W1, const float* __restrict__ b1,
                                const float* __restrict__ W2, const float* __restrict__ b2,
                                float* __restrict__ f) {
  extern __shared__ float sm[];
  float* xs  = sm;                 // 128 x 36 (K padded 35->36)
  float* ys  = xs  + 128 * 36;     // 128 x 64 relu intermediate
  float* w1s = ys  + 128 * 64;     // 36 x 64 (row 35 zero)
  float* w2s = w1s + 36 * 64;      // 64 x 64
  float* b1s = w2s + 64 * 64;      // 64
  float* b2s = b1s + 64;           // 64
  const int t    = threadIdx.x;    // 0..255
  const int rowg = blockIdx.x * 128;

  for (int i = t; i < 36 * 64; i += 256) {
    int k = i >> 6;
    w1s[i] = (k < 35) ? W1[i] : 0.0f;
  }
  for (int i = t; i < 64 * 64; i += 256) w2s[i] = W2[i];
  if (t < 64) { b1s[t] = b1[t]; b2s[t] = b2[t]; }
  for (int i = t; i < 128 * 36; i += 256) {
    int r = i / 36, k = i - r * 36;
    int p = rowg + r;
    float v;
    if (k < 32)      v = h[(size_t)p * C_ + k];
    else if (k < 35) v = pos[(size_t)p * 3 + (k - 32)];
    else             v = 0.0f;
    xs[i] = v;
  }
  __syncthreads();

  const int wave = t >> 5, lane = t & 31;
  const int half = lane >> 4, r = lane & 15;
  const int rloc = wave * 16;

  // Stage 1: y = relu(x @ W1 + b1), K = 36 (9 steps of 4)
  v8f acc[4] = {};
  #pragma unroll
  for (int k0 = 0; k0 < 9; ++k0) {
    const int ka = k0 * 4 + 2 * half;
    v2f a;
    a.x = xs[(rloc + r) * 36 + ka];
    a.y = xs[(rloc + r) * 36 + ka + 1];
    #pragma unroll
    for (int nt = 0; nt < 4; ++nt) {
      v2f bb;
      bb.x = w1s[ka * 64 + nt * 16 + r];
      bb.y = w1s[(ka + 1) * 64 + nt * 16 + r];
      acc[nt] = __builtin_amdgcn_wmma_f32_16x16x4_f32(
          false, a, false, bb, (short)0, acc[nt], false, false);
    }
  }
  // bias + relu, round-trip through LDS to convert C-layout -> A-layout
  #pragma unroll
  for (int nt = 0; nt < 4; ++nt) {
    #pragma unroll
    for (int v = 0; v < 8; ++v) {
      int rr = rloc + v + 8 * half;
      ys[rr * 64 + nt * 16 + r] = fmaxf(acc[nt][v] + b1s[nt * 16 + r], 0.0f);
    }
  }
  __syncthreads();

  // Stage 2: z = y @ W2 + b2, K = 64 (16 steps of 4)
  v8f acc2[4] = {};
  #pragma unroll
  for (int k0 = 0; k0 < 16; ++k0) {
    const int ka = k0 * 4 + 2 * half;
    v2f a;
    a.x = ys[(rloc + r) * 64 + ka];
    a.y = ys[(rloc + r) * 64 + ka + 1];
    #pragma unroll
    for (int nt = 0; nt < 4; ++nt) {
      v2f bb;
      bb.x = w2s[ka * 64 + nt * 16 + r];
      bb.y = w2s[(ka + 1) * 64 + nt * 16 + r];
      acc2[nt] = __builtin_amdgcn_wmma_f32_16x16x4_f32(
          false, a, false, bb, (short)0, acc2[nt], false, false);
    }
  }
  #pragma unroll
  for (int nt = 0; nt < 4; ++nt) {
    #pragma unroll
    for (int v = 0; v < 8; ++v) {
      int rr = rloc + v + 8 * half;
      int p  = rowg + rr;
      f[(size_t)p * OUT_ + nt * 16 + r] = acc2[nt][v] + b2s[nt * 16 + r];
    }
  }
}

// ---------------------------------------------------------------------------
// Kernel 3: radius search + exact top-64-by-d2. One thread per centroid;
// per-thread 64-slot (d2, idx) lists live in LDS, strided by tid.
// ---------------------------------------------------------------------------
__global__ void nbr_kernel(const float* __restrict__ pos, const float* __restrict__ cent,
                           int* __restrict__ nidx, int* __restrict__ ncnt) {
  extern __shared__ float sm[];
  float* px  = sm;
  float* py  = px + N_;
  float* pz  = py + N_;
  float* d2l = pz + N_;                 // 64 * 128
  int*   il  = (int*)(d2l + K_ * 128);  // 64 * 128
  const int t  = threadIdx.x;           // 0..127
  const int b  = blockIdx.x >> 3;
  const int c  = ((blockIdx.x & 7) << 7) + t;
  const int gc = b * S_ + c;
  const float* pb = pos + (size_t)b * N_ * 3;

  for (int i = t; i < N_; i += 128) {
    px[i] = pb[i * 3 + 0];
    py[i] = pb[i * 3 + 1];
    pz[i] = pb[i * 3 + 2];
  }
  __syncthreads();

  const float cx = cent[gc * 3 + 0], cy = cent[gc * 3 + 1], cz = cent[gc * 3 + 2];
  int cnt = 0; float wmax = -1.0f; int wslot = 0;
  for (int j = 0; j < N_; ++j) {
    float dx = px[j] - cx, dy = py[j] - cy, dz = pz[j] - cz;
    float d2 = dx * dx + dy * dy + dz * dz;
    if (d2 <= R2_) {
      if (cnt < K_) {
        d2l[cnt * 128 + t] = d2;
        il[cnt * 128 + t]  = j;
        if (d2 > wmax) { wmax = d2; wslot = cnt; }
        ++cnt;
      } else if (d2 < wmax) {     // strict-less keeps earliest index on ties
        d2l[wslot * 128 + t] = d2;
        il[wslot * 128 + t]  = j;
        wmax = -1.0f;
        for (int k = 0; k < K_; ++k) {
          float v = d2l[k * 128 + t];
          if (v > wmax) { wmax = v; wslot = k; }
        }
      }
    }
  }
  ncnt[gc] = cnt;
  for (int k = 0; k < cnt; ++k) nidx[(size_t)gc * K_ + k] = il[k * 128 + t];
}

// ---------------------------------------------------------------------------
// Kernel 4: new_h[c, ch] = max over neighbors of f[j, ch] (0 if none).
// 256 thr = 4 centroids x 64 channels (coalesced feature reads).
// ---------------------------------------------------------------------------
__global__ void agg_kernel(const int* __restrict__ nidx, const int* __restrict__ ncnt,
                           const float* __restrict__ f, float* __restrict__ newh) {
  const int gc = blockIdx.x * 4 + (threadIdx.x >> 6);
  const int ch = threadIdx.x & 63;
  const int b  = gc >> 10;
  const int cnt = ncnt[gc];
  const int* lst = nidx + (size_t)gc * K_;
  float m = -__builtin_inff();
  for (int i = 0; i < cnt; ++i) {
    int j = lst[i];
    m = fmaxf(m, f[((size_t)(b * N_ + j)) * OUT_ + ch]);
  }
  newh[(size_t)gc * OUT_ + ch] = (cnt > 0) ? m : 0.0f;
}

// ---------------------------------------------------------------------------
extern "C" void kernel_launch(void* const* d_in, const int* in_sizes, int n_in,
                              void* d_out, int out_size, void* d_ws, size_t ws_size,
                              hipStream_t stream) {
  (void)in_sizes; (void)n_in; (void)out_size; (void)ws_size;
  const float* pos = (const float*)d_in[0];
  const float* h   = (const float*)d_in[1];
  const float* W1  = (const float*)d_in[2];
  const float* b1  = (const float*)d_in[3];
  const float* W2  = (const float*)d_in[4];
  const float* b2  = (const float*)d_in[5];

  float* out_cent = (float*)d_out;               // B*S*3 = 24576 floats
  float* out_newh = out_cent + (size_t)B_ * S_ * 3;

  char* ws = (char*)d_ws;
  float* f    = (float*)ws;                                  // 32768*64 f32 (8.39MB)
  int*   nidx = (int*)(ws + (size_t)B_ * N_ * OUT_ * 4);     // 8192*64 int (2MB)
  int*   ncnt = (int*)(ws + (size_t)B_ * N_ * OUT_ * 4
                          + (size_t)B_ * S_ * K_ * 4);       // 8192 int

  const size_t fps_smem = (size_t)(4 * N_ + 64) * 4;                 // 65,792 B
  const size_t mlp_smem = (size_t)(128*36 + 128*64 + 36*64 + 64*64 + 128) * 4; // 77,312 B
  const size_t nbr_smem = (size_t)(3 * N_ + K_ * 128) * 4 + (size_t)K_ * 128 * 4; // 114,688 B

  fps_kernel<<<B_, 1024, fps_smem, stream>>>(pos, out_cent);
  mlp_wmma_kernel<<<(B_ * N_) / 128, 256, mlp_smem, stream>>>(h, pos, W1, b1, W2, b2, f);
  nbr_kernel<<<(B_ * S_) / 128, 128, nbr_smem, stream>>>(pos, out_cent, nidx, ncnt);
  agg_kernel<<<(B_ * S_) / 4, 256, 0, stream>>>(nidx, ncnt, f, out_newh);
}